// DecoupledGeometricEncoder_77137612636234
// MI455X (gfx1250) — compile-verified
//
#include <hip/hip_runtime.h>
#include <hip/hip_bf16.h>

typedef __attribute__((ext_vector_type(2))) float v2f;
typedef __attribute__((ext_vector_type(8))) float v8f;

#define BATCH 8
#define NPTS  4096
#define KNN   5
#define EPSV  1e-6f
#define SLAB_W 132   // 16-row slab stride: 132 % 64 banks -> stride-4 bank walk, conflict free

// ---------------------------------------------------------------------------
// Kernel 0: zero-pad the two ragged weight matrices so every WMMA B-fragment
// load is unconditional (no divergent exec-mask dances around global loads).
//   lw1 (2,64)  -> (4,64)   ; cw1 (66,128) -> (68,128)
// ---------------------------------------------------------------------------
__global__ __launch_bounds__(256) void pad_weights_kernel(
    const float* __restrict__ lw1, float* __restrict__ plw1,
    const float* __restrict__ cw1, float* __restrict__ pcw1)
{
    const int i = blockIdx.x * 256 + threadIdx.x;
    if (i < 4 * 64)   plw1[i] = (i < 2 * 64)   ? lw1[i] : 0.f;
    if (i < 68 * 128) pcw1[i] = (i < 66 * 128) ? cw1[i] : 0.f;
}

// ---------------------------------------------------------------------------
// Kernel 1: brute-force KNN (top-5) + 2x2 covariance eigendecomposition.
// One block = 256 query points of one batch; whole batch point cloud in LDS
// as interleaved float2 (one broadcast ds_load_b64 per candidate).
// ---------------------------------------------------------------------------
__global__ __launch_bounds__(256) void knn_geom_kernel(
    const float* __restrict__ x,
    float* __restrict__ lcc,        // (B*N*K, 2)
    float* __restrict__ out_angle,  // (B*N)
    float* __restrict__ out_axes)   // (B*N, 2)
{
    __shared__ float2 sp[NPTS];

    const int b = blockIdx.x >> 4;                       // NPTS/256 = 16 blocks per batch
    const int n = ((blockIdx.x & 15) << 8) + threadIdx.x;
    const float2* xb = (const float2*)(x + (size_t)b * NPTS * 2);

    for (int i = threadIdx.x; i < NPTS; i += 256) sp[i] = xb[i];
    __syncthreads();

    const float2 q = sp[n];

    float qd0 = 3.4e38f, qd1 = 3.4e38f, qd2 = 3.4e38f, qd3 = 3.4e38f, qd4 = 3.4e38f;
    int   qi0 = 0, qi1 = 0, qi2 = 0, qi3 = 0, qi4 = 0;

#pragma unroll 4
    for (int m = 0; m < NPTS; ++m) {
        const float2 p2 = sp[m];
        const float dx = p2.x - q.x;
        const float dy = p2.y - q.y;
        const float d = fmaf(dx, dx, dy * dy);
        if (d < qd4) {
            if (d < qd3) {
                qd4 = qd3; qi4 = qi3;
                if (d < qd2) {
                    qd3 = qd2; qi3 = qi2;
                    if (d < qd1) {
                        qd2 = qd1; qi2 = qi1;
                        if (d < qd0) { qd1 = qd0; qi1 = qi0; qd0 = d; qi0 = m; }
                        else         { qd1 = d;   qi1 = m; }
                    } else { qd2 = d; qi2 = m; }
                } else { qd3 = d; qi3 = m; }
            } else { qd4 = d; qi4 = m; }
        }
    }

    const int qi[KNN] = { qi0, qi1, qi2, qi3, qi4 };
    float rx[KNN], ry[KNN];
#pragma unroll
    for (int k = 0; k < KNN; ++k) {
        const float2 p2 = sp[qi[k]];
        rx[k] = p2.x - q.x; ry[k] = p2.y - q.y;
    }

    const float mx = (rx[0] + rx[1] + rx[2] + rx[3] + rx[4]) * 0.2f;
    const float my = (ry[0] + ry[1] + ry[2] + ry[3] + ry[4]) * 0.2f;

    float cx[KNN], cy[KNN];
    float ca = 0.f, cb = 0.f, cc = 0.f;
#pragma unroll
    for (int k = 0; k < KNN; ++k) {
        cx[k] = rx[k] - mx; cy[k] = ry[k] - my;
        ca = fmaf(cx[k], cx[k], ca);
        cb = fmaf(cx[k], cy[k], cb);
        cc = fmaf(cy[k], cy[k], cc);
    }
    ca = ca * 0.25f + EPSV;   // / (K-1) + EPS*I
    cb = cb * 0.25f;
    cc = cc * 0.25f + EPSV;

    // closed-form symmetric 2x2 eigendecomposition (ascending e0 <= e1)
    const float ht = 0.5f * (ca + cc);
    const float df = 0.5f * (ca - cc);
    const float disc = sqrtf(fmaf(df, df, cb * cb));
    const float e1 = ht + disc;
    const float e0 = ht - disc;

    // eigenvector for e1: pick the better-conditioned candidate
    float ax1 = cb,      ay1 = e1 - ca;
    float ax2 = e1 - cc, ay2 = cb;
    const float n1 = fmaf(ax1, ax1, ay1 * ay1);
    const float n2 = fmaf(ax2, ax2, ay2 * ay2);
    float v1x = (n1 >= n2) ? ax1 : ax2;
    float v1y = (n1 >= n2) ? ay1 : ay2;
    float nn = fmaxf(n1, n2);
    if (nn < 1e-30f) { v1x = 1.f; v1y = 0.f; nn = 1.f; }
    const float inv = rsqrtf(nn);
    v1x *= inv; v1y *= inv;
    const float v0x = -v1y, v0y = v1x;

    const int p = b * NPTS + n;
    out_angle[p] = atan2f(v1y, v1x);

    const float s1 = sqrtf(fmaxf(e1, EPSV));
    const float s0 = sqrtf(fmaxf(e0, EPSV));
    const float dm = 1.0f / (s1 + EPSV);
    out_axes[2 * p]     = fmaxf(s1 * dm, 0.2f);
    out_axes[2 * p + 1] = fmaxf(s0 * dm, 0.2f);

#pragma unroll
    for (int k = 0; k < KNN; ++k) {
        const size_t o = ((size_t)p * KNN + k) * 2;
        lcc[o]     = fmaf(cx[k], v0x, cy[k] * v0y);
        lcc[o + 1] = fmaf(cx[k], v1x, cy[k] * v1y);
    }
}

// ---------------------------------------------------------------------------
// Dense layer on one wave: out = relu(in @ W + bias), 16-row tile,
// V_WMMA_F32_16X16X4_F32 (exact fp32). KPAD multiple of 4; W has KPAD rows
// (pre-padded with zeros where needed), so all loads are unconditional.
// k0 outer loop: one A fragment feeds all NT column-tile WMMAs (operand reuse,
// fewer LDS reads); all NT accumulators live in VGPRs.
//   A frag: lane l holds A[l%16][k0 + (l/16)*2 + r]   (ISA 7.12.2, 32-bit A 16x4)
//   B frag: lane l holds B[k0 + (l/16)*2 + r][l%16]
//   C/D   : lane l holds D[r + (l/16)*8][l%16]
// ---------------------------------------------------------------------------
template<int KPAD, int NOUT>
__device__ __forceinline__ void wmma_layer(
    const float* sin_, float* sout,
    const float* __restrict__ W, const float* __restrict__ bias, int lane)
{
    constexpr int NT = NOUT / 16;
    const int col   = lane & 15;
    const int khalf = (lane >> 4) * 2;
    const int dhalf = (lane >> 4) * 8;

    v8f c[NT];
#pragma unroll
    for (int j = 0; j < NT; ++j) {
        const float bv = bias[j * 16 + col];
#pragma unroll
        for (int r = 0; r < 8; ++r) c[j][r] = bv;
    }

#pragma unroll
    for (int k0 = 0; k0 < KPAD; k0 += 4) {
        const int ka = k0 + khalf;
        v2f a;
        a[0] = sin_[col * SLAB_W + ka];
        a[1] = sin_[col * SLAB_W + ka + 1];
#pragma unroll
        for (int j = 0; j < NT; ++j) {
            v2f bm;
            bm[0] = W[(size_t)ka * NOUT + j * 16 + col];
            bm[1] = W[(size_t)(ka + 1) * NOUT + j * 16 + col];
            c[j] = __builtin_amdgcn_wmma_f32_16x16x4_f32(false, a, false, bm,
                                                         (short)0, c[j], false, false);
        }
    }

#pragma unroll
    for (int j = 0; j < NT; ++j)
#pragma unroll
        for (int r = 0; r < 8; ++r)
            sout[(r + dhalf) * SLAB_W + j * 16 + col] = fmaxf(c[j][r], 0.f);
}

// ---------------------------------------------------------------------------
// Kernel 2: per-neighbor MLP 2->64->128->64 over 163840 rows. 2 waves/block,
// 16 rows/wave, activations ping-pong through padded LDS slabs.
// ---------------------------------------------------------------------------
__global__ __launch_bounds__(64) void local_mlp_kernel(
    const float* __restrict__ lcc, float* __restrict__ h3,
    const float* __restrict__ plw1, const float* __restrict__ lb1,
    const float* __restrict__ lw2,  const float* __restrict__ lb2,
    const float* __restrict__ lw3,  const float* __restrict__ lb3)
{
    __shared__ float slabA[2][16 * SLAB_W];
    __shared__ float slabB[2][16 * SLAB_W];

    const int wave = threadIdx.x >> 5;
    const int lane = threadIdx.x & 31;
    const int row_base = blockIdx.x * 32 + wave * 16;
    float* A  = slabA[wave];
    float* Bs = slabB[wave];

    __builtin_prefetch(lw2, 0, 3);
    __builtin_prefetch(lw3, 0, 3);

    // stage lcc tile, padded K: cols 2..3 = 0
    for (int i = lane; i < 16 * 4; i += 32) {
        const int r = i >> 2, cdx = i & 3;
        A[r * SLAB_W + cdx] = (cdx < 2) ? lcc[(size_t)(row_base + r) * 2 + cdx] : 0.f;
    }
    __syncthreads();
    wmma_layer<4, 64>(A, Bs, plw1, lb1, lane);
    __syncthreads();
    wmma_layer<64, 128>(Bs, A, lw2, lb2, lane);
    __syncthreads();
    wmma_layer<128, 64>(A, Bs, lw3, lb3, lane);
    __syncthreads();
    for (int i = lane; i < 16 * 64; i += 32) {
        const int r = i >> 6, cdx = i & 63;
        h3[(size_t)(row_base + r) * 64 + cdx] = Bs[r * SLAB_W + cdx];
    }
}

// ---------------------------------------------------------------------------
// Kernel 3: local_feats = max over K neighbors
// ---------------------------------------------------------------------------
__global__ __launch_bounds__(256) void maxk_kernel(
    const float* __restrict__ h3, float* __restrict__ lf)
{
    const int idx = blockIdx.x * 256 + threadIdx.x;
    if (idx >= BATCH * NPTS * 64) return;
    const int p = idx >> 6, c = idx & 63;
    const float* base = h3 + ((size_t)p * KNN) * 64 + c;
    float m = base[0];
    m = fmaxf(m, base[64]);
    m = fmaxf(m, base[128]);
    m = fmaxf(m, base[192]);
    m = fmaxf(m, base[256]);
    lf[idx] = m;
}

// ---------------------------------------------------------------------------
// Kernel 4: cls head (concat(x, lf): 66(+pad 68)->128->128) +
//           topo head (64->128->128)
// ---------------------------------------------------------------------------
__global__ __launch_bounds__(64) void head_mlp_kernel(
    const float* __restrict__ x, const float* __restrict__ lf,
    float* __restrict__ out_cls, float* __restrict__ out_topo,
    const float* __restrict__ pcw1, const float* __restrict__ cb1,
    const float* __restrict__ cw2,  const float* __restrict__ cb2,
    const float* __restrict__ tw1,  const float* __restrict__ tb1,
    const float* __restrict__ tw2,  const float* __restrict__ tb2)
{
    __shared__ float slabA[2][16 * SLAB_W];
    __shared__ float slabB[2][16 * SLAB_W];

    const int wave = threadIdx.x >> 5;
    const int lane = threadIdx.x & 31;
    const int row_base = blockIdx.x * 32 + wave * 16;
    float* A  = slabA[wave];
    float* Bs = slabB[wave];

    __builtin_prefetch(pcw1, 0, 3);
    __builtin_prefetch(cw2, 0, 3);
    __builtin_prefetch(tw1, 0, 3);
    __builtin_prefetch(tw2, 0, 3);

    // cls_in = [x(2) | lf(64) | pad(2)] -> 68 cols
    for (int i = lane; i < 16 * 68; i += 32) {
        const int r = i / 68, cdx = i % 68;
        float v;
        if (cdx < 2)       v = x[(size_t)(row_base + r) * 2 + cdx];
        else if (cdx < 66) v = lf[(size_t)(row_base + r) * 64 + (cdx - 2)];
        else               v = 0.f;
        A[r * SLAB_W + cdx] = v;
    }
    __syncthreads();
    wmma_layer<68, 128>(A, Bs, pcw1, cb1, lane);
    __syncthreads();
    wmma_layer<128, 128>(Bs, A, cw2, cb2, lane);
    __syncthreads();
    for (int i = lane; i < 16 * 128; i += 32) {
        const int r = i >> 7, cdx = i & 127;
        out_cls[(size_t)(row_base + r) * 128 + cdx] = A[r * SLAB_W + cdx];
    }
    __syncthreads();

    // topo path
    for (int i = lane; i < 16 * 64; i += 32) {
        const int r = i >> 6, cdx = i & 63;
        A[r * SLAB_W + cdx] = lf[(size_t)(row_base + r) * 64 + cdx];
    }
    __syncthreads();
    wmma_layer<64, 128>(A, Bs, tw1, tb1, lane);
    __syncthreads();
    wmma_layer<128, 128>(Bs, A, tw2, tb2, lane);
    __syncthreads();
    for (int i = lane; i < 16 * 128; i += 32) {
        const int r = i >> 7, cdx = i & 127;
        out_topo[(size_t)(row_base + r) * 128 + cdx] = A[r * SLAB_W + cdx];
    }
}

// ---------------------------------------------------------------------------
extern "C" void kernel_launch(void* const* d_in, const int* in_sizes, int n_in,
                              void* d_out, int out_size, void* d_ws, size_t ws_size,
                              hipStream_t stream) {
    const float* x   = (const float*)d_in[0];
    const float* lw1 = (const float*)d_in[1];
    const float* lb1 = (const float*)d_in[2];
    const float* lw2 = (const float*)d_in[3];
    const float* lb2 = (const float*)d_in[4];
    const float* lw3 = (const float*)d_in[5];
    const float* lb3 = (const float*)d_in[6];
    const float* cw1 = (const float*)d_in[7];
    const float* cb1 = (const float*)d_in[8];
    const float* cw2 = (const float*)d_in[9];
    const float* cb2 = (const float*)d_in[10];
    const float* tw1 = (const float*)d_in[11];
    const float* tb1 = (const float*)d_in[12];
    const float* tw2 = (const float*)d_in[13];
    const float* tb2 = (const float*)d_in[14];

    const int P = BATCH * NPTS;            // 32768 points
    const int R = P * KNN;                 // 163840 neighbor rows

    float* out       = (float*)d_out;
    float* out_cls   = out;                          // P*128
    float* out_topo  = out + (size_t)P * 128;        // P*128
    float* out_angle = out_topo + (size_t)P * 128;   // P
    float* out_axes  = out_angle + P;                // P*2

    float* w       = (float*)d_ws;
    float* ws_lcc  = w;                              // R*2
    float* ws_h3   = ws_lcc + (size_t)R * 2;         // R*64
    float* ws_lf   = ws_h3 + (size_t)R * 64;         // P*64
    float* ws_plw1 = ws_lf + (size_t)P * 64;         // 4*64
    float* ws_pcw1 = ws_plw1 + 4 * 64;               // 68*128

    pad_weights_kernel<<<(68 * 128 + 255) / 256, 256, 0, stream>>>(
        lw1, ws_plw1, cw1, ws_pcw1);

    knn_geom_kernel<<<BATCH * (NPTS / 256), 256, 0, stream>>>(
        x, ws_lcc, out_angle, out_axes);

    local_mlp_kernel<<<R / 32, 64, 0, stream>>>(
        ws_lcc, ws_h3, ws_plw1, lb1, lw2, lb2, lw3, lb3);

    maxk_kernel<<<(P * 64) / 256, 256, 0, stream>>>(ws_h3, ws_lf);

    head_mlp_kernel<<<P / 32, 64, 0, stream>>>(
        x, ws_lf, out_cls, out_topo,
        ws_pcw1, cb1, cw2, cb2, tw1, tb1, tw2, tb2);
}